// SignalLoss_63720134803972
// MI455X (gfx1250) — compile-verified
//
#include <hip/hip_runtime.h>
#include <math.h>

// Problem constants (match reference)
#define BATCH 256
#define LEN   65536
#define NB    4                    // partial blocks per row == WMMA K dimension
#define CHUNK (LEN / NB)           // 16384 floats per (row, block)
#define CHUNK4 (CHUNK / 4)         // 4096 float4 per (row, block)
#define T1    256                  // stage-1 block size (8 wave32s)
#define MSE_W 0.7f
#define CORR_W 0.3f
#define EPS_F 1e-8f

typedef __attribute__((ext_vector_type(2))) float v2f;
typedef __attribute__((ext_vector_type(4))) float v4f;
typedef __attribute__((ext_vector_type(8))) float v8f;

__device__ __forceinline__ float wave_sum32(float v) {
#pragma unroll
    for (int off = 16; off > 0; off >>= 1)
        v += __shfl_xor(v, off, 32);
    return v;
}

// ---------------------------------------------------------------------------
// Stage 1: streaming pass. Grid (NB, BATCH). Each block reduces a 16384-elem
// slice of one row into 5 f32 stats: [sx, sy, sxx, syy, sxy] (x=true, y=pred).
// Bandwidth-bound: b128 non-temporal loads, f32 FMA accumulation.
// ---------------------------------------------------------------------------
__global__ __launch_bounds__(T1) void signal_loss_stage1(
    const float* __restrict__ pred, const float* __restrict__ truev,
    float* __restrict__ P /* [BATCH][NB][5] */) {
    const int row = blockIdx.y;
    const int blk = blockIdx.x;
    const size_t base = (size_t)row * LEN + (size_t)blk * CHUNK;
    const v4f* __restrict__ p4 = (const v4f*)(pred + base);
    const v4f* __restrict__ t4 = (const v4f*)(truev + base);

    float sx = 0.f, sy = 0.f, sxx = 0.f, syy = 0.f, sxy = 0.f;
#pragma unroll 4
    for (int i = threadIdx.x; i < CHUNK4; i += T1) {
        v4f p = __builtin_nontemporal_load(&p4[i]);
        v4f t = __builtin_nontemporal_load(&t4[i]);
#pragma unroll
        for (int c = 0; c < 4; ++c) {
            float tc = t[c], pc = p[c];
            sx += tc;
            sy += pc;
            sxx = fmaf(tc, tc, sxx);
            syy = fmaf(pc, pc, syy);
            sxy = fmaf(tc, pc, sxy);
        }
    }

    sx  = wave_sum32(sx);
    sy  = wave_sum32(sy);
    sxx = wave_sum32(sxx);
    syy = wave_sum32(syy);
    sxy = wave_sum32(sxy);

    __shared__ float lds[T1 / 32][5];
    const int lane = threadIdx.x & 31;
    const int wave = threadIdx.x >> 5;
    if (lane == 0) {
        lds[wave][0] = sx; lds[wave][1] = sy; lds[wave][2] = sxx;
        lds[wave][3] = syy; lds[wave][4] = sxy;
    }
    __syncthreads();
    if (threadIdx.x < 5) {
        float acc = 0.f;
#pragma unroll
        for (int w = 0; w < T1 / 32; ++w) acc += lds[w][threadIdx.x];
        P[((size_t)row * NB + blk) * 5 + threadIdx.x] = acc;
    }
}

// ---------------------------------------------------------------------------
// Stage 2: one block, 16 wave32s. Wave w owns rows [16w, 16w+16). The per-row
// reduction over NB=4 partials is a genuine 16x4 (f32) x 4x16 (ones) matmul:
// D = A * ones + 0, so D[m,n] = sum_k A[m,k]  -> full-precision f32 WMMA.
//   A layout (ISA 7.12.2, 32-bit A 16x4): lane = M + 16*(K>=2), VGPR = K&1.
//   B = all ones (layout-invariant).
//   D layout (32-bit C/D 16x16): VGPR r = row M=r (lanes 0-15) / M=r+8 (16-31).
// ---------------------------------------------------------------------------
__global__ __launch_bounds__(512) void signal_loss_stage2(
    const float* __restrict__ P, float* __restrict__ out) {
    const int tid  = threadIdx.x;
    const int lane = tid & 31;
    const int wave = tid >> 5;
    const int row0 = wave * 16;
    const int m    = lane & 15;          // A-matrix row owned by this lane
    const int koff = (lane >> 4) << 1;   // lanes 16-31 carry K=2,3

    const v2f bones = {1.0f, 1.0f};
    float stat[5];
#pragma unroll
    for (int s = 0; s < 5; ++s) {
        v2f a;
        a.x = P[((size_t)(row0 + m) * NB + koff + 0) * 5 + s];
        a.y = P[((size_t)(row0 + m) * NB + koff + 1) * 5 + s];
        v8f c = {};
        // D[m,n] = sum_{k=0..3} P[row0+m][k][s]   (replicated over n)
        v8f d = __builtin_amdgcn_wmma_f32_16x16x4_f32(
            false, a, false, bones, (short)0, c, false, false);
        // lane holds rows M = r (+8 if lane>=16) in d[r]; extract row (lane&7 group)
        float val;
        switch (lane & 7) {
            case 0: val = d[0]; break;
            case 1: val = d[1]; break;
            case 2: val = d[2]; break;
            case 3: val = d[3]; break;
            case 4: val = d[4]; break;
            case 5: val = d[5]; break;
            case 6: val = d[6]; break;
            default: val = d[7]; break;
        }
        // redistribute so lane l holds row (l & 15):
        // row m<8 lives in lanes 0-15 at d[m]; row m>=8 lives in lanes 16-31 at d[m-8]
        int src = (m < 8) ? m : (m + 8);
        stat[s] = __shfl(val, src, 32);
    }

    const float invL = 1.0f / (float)LEN;
    float sx = stat[0], sy = stat[1], sxx = stat[2], syy = stat[3], sxy = stat[4];
    float rnum = sxy - sx * sy * invL;
    float varx = sxx - sx * sx * invL;
    float vary = syy - sy * sy * invL;
    float den  = fmaxf(sqrtf(varx * vary), EPS_F);
    float term = 1.0f - rnum / den;                 // per-row (1 - r)
    float rowsq = sxx + syy - 2.0f * sxy;           // per-row sum (pred-true)^2

    // lanes 16-31 are duplicates of 0-15; xor offsets < 16 keep groups separate
#pragma unroll
    for (int off = 8; off > 0; off >>= 1) {
        term  += __shfl_xor(term, off, 32);
        rowsq += __shfl_xor(rowsq, off, 32);
    }

    __shared__ float ldsT[16];
    __shared__ float ldsQ[16];
    if (lane == 0) { ldsT[wave] = term; ldsQ[wave] = rowsq; }
    __syncthreads();
    if (wave == 0) {
        float t = ldsT[lane & 15];
        float q = ldsQ[lane & 15];
#pragma unroll
        for (int off = 8; off > 0; off >>= 1) {
            t += __shfl_xor(t, off, 32);
            q += __shfl_xor(q, off, 32);
        }
        if (lane == 0) {
            float mse  = q / (float)((size_t)BATCH * (size_t)LEN); // 2^24, exact
            float corr = t / (float)BATCH;
            out[0] = MSE_W * mse + CORR_W * corr;
        }
    }
}

extern "C" void kernel_launch(void* const* d_in, const int* in_sizes, int n_in,
                              void* d_out, int out_size, void* d_ws, size_t ws_size,
                              hipStream_t stream) {
    (void)in_sizes; (void)n_in; (void)out_size; (void)ws_size;
    const float* y_pred = (const float*)d_in[0];
    const float* y_true = (const float*)d_in[1];
    float* P = (float*)d_ws;               // [BATCH][NB][5] = 20 KiB scratch

    dim3 grid1(NB, BATCH);
    signal_loss_stage1<<<grid1, T1, 0, stream>>>(y_pred, y_true, P);
    signal_loss_stage2<<<1, 512, 0, stream>>>(P, (float*)d_out);
}